// LocalMeanInpainter_35424890257513
// MI455X (gfx1250) — compile-verified
//
#include <hip/hip_runtime.h>
#include <stdint.h>

#define BATCH     32
#define CHANNELS  3
#define HEIGHT    512
#define WIDTH     512
#define WINDOW    15
#define PADR      7
#define TILE      64
#define HALO      (TILE + WINDOW - 1)   // 78 valid rows/cols in staged tile
#define HROWS     80                    // padded rows (K reach of WMMA chain)
#define HCOLS     80                    // padded cols (N reach of WMMA chunks)
#define PITCH     81                    // bank-conflict-free row pitch
#define OPITCH    (TILE + 1)            // 65
#define NTHREADS  256

#ifndef __has_builtin
#define __has_builtin(x) 0
#endif

#if __has_builtin(__builtin_amdgcn_global_load_async_to_lds_b32)
#define HAVE_ASYNC 1
#else
#define HAVE_ASYNC 0
#endif

typedef __attribute__((address_space(1))) int* gint_p;
typedef __attribute__((address_space(3))) int* lint_p;
typedef __attribute__((ext_vector_type(2))) float v2f;
typedef __attribute__((ext_vector_type(8))) float v8f;

// Copy 4 bytes global -> LDS via the CDNA5 async-to-LDS path (ASYNCcnt).
__device__ __forceinline__ void g2l_b32(const float* g, float* l) {
#if HAVE_ASYNC
  __builtin_amdgcn_global_load_async_to_lds_b32(
      (gint_p)(uintptr_t)g,
      (lint_p)(uint32_t)(uintptr_t)l,
      0, 0);
#else
  *l = *g;
#endif
}

__device__ __forceinline__ void wait_async_all() {
#if HAVE_ASYNC
#if __has_builtin(__builtin_amdgcn_s_wait_asynccnt)
  __builtin_amdgcn_s_wait_asynccnt(0);
#else
  asm volatile("s_wait_asynccnt 0" ::: "memory");
#endif
#endif
}

__global__ __launch_bounds__(NTHREADS)
void inpaint_kernel(const float* __restrict__ x,
                    const float* __restrict__ mask,
                    float* __restrict__ out) {
  const int tx0 = blockIdx.x * TILE;
  const int ty0 = blockIdx.y * TILE;
  const int b   = blockIdx.z;
  const int tid = threadIdx.x;

  __shared__ float s_mask[HROWS * PITCH];      // mask halo tile (persistent)
  __shared__ float s_x[2][HROWS * PITCH];      // double-buffered x halo tile
  __shared__ float s_v[TILE * PITCH];          // vertical sums of x*mask
  __shared__ float s_vn[TILE * PITCH];         // vertical sums of mask
  __shared__ float s_out[TILE * OPITCH];       // staged output (coalesced store)

  const float* mbase = mask + (size_t)b * HEIGHT * WIDTH;
  const float* xbase = x    + (size_t)b * CHANNELS * HEIGHT * WIDTH;
  float*       obase = out  + (size_t)b * CHANNELS * HEIGHT * WIDTH;

  // ---- issue async loads: mask tile + channel-0 tile (zero pad + OOB) ----
  for (int i = tid; i < HROWS * HCOLS; i += NTHREADS) {
    int r = i / HCOLS, c = i - r * HCOLS;
    int gr = ty0 - PADR + r;
    int gc = tx0 - PADR + c;
    bool ok = (r < HALO) && (c < HALO) &&
              ((unsigned)gr < HEIGHT) && ((unsigned)gc < WIDTH);
    float* dm = &s_mask[r * PITCH + c];
    float* dx = &s_x[0][r * PITCH + c];
    if (ok) {
      g2l_b32(mbase + gr * WIDTH + gc, dm);
      g2l_b32(xbase + gr * WIDTH + gc, dx);   // channel 0
    } else {
      *dm = 0.0f;
      *dx = 0.0f;
    }
  }
  wait_async_all();
  __syncthreads();

  // ---- vertical sliding-window sums of mask -> s_vn (computed once) ----
  for (int t = tid; t < 4 * HALO; t += NTHREADS) {
    int c  = t % HALO;
    int rb = (t / HALO) * 16;
    float sum = 0.0f;
#pragma unroll
    for (int k = 0; k < WINDOW; ++k) sum += s_mask[(rb + k) * PITCH + c];
#pragma unroll
    for (int j = 0; j < 16; ++j) {
      s_vn[(rb + j) * PITCH + c] = sum;
      if (j < 15)
        sum += s_mask[(rb + j + WINDOW) * PITCH + c]
             - s_mask[(rb + j) * PITCH + c];
    }
  }
  // s_vn consumed only after the barrier inside the channel loop

  const int wave = tid >> 5;        // 8 waves of 32 (wave32)
  const int lane = tid & 31;
  const int half = lane >> 4;       // 0: lanes 0-15, 1: lanes 16-31
  const int lm   = lane & 15;       // M (A) / N (B,D) index within 16

  for (int ch = 0; ch < CHANNELS; ++ch) {
    const int cur = ch & 1;

    // ---- prefetch next channel into the other buffer (async, no wait) ----
    if (ch + 1 < CHANNELS) {
      const int nxt = cur ^ 1;
      const float* xc = xbase + (size_t)(ch + 1) * HEIGHT * WIDTH;
      for (int i = tid; i < HROWS * HCOLS; i += NTHREADS) {
        int r = i / HCOLS, c = i - r * HCOLS;
        int gr = ty0 - PADR + r;
        int gc = tx0 - PADR + c;
        bool ok = (r < HALO) && (c < HALO) &&
                  ((unsigned)gr < HEIGHT) && ((unsigned)gc < WIDTH);
        float* dx = &s_x[nxt][r * PITCH + c];
        if (ok) g2l_b32(xc + gr * WIDTH + gc, dx);
        else    *dx = 0.0f;
      }
    }

    // ---- vertical 15-tap sums of x*mask -> s_v via banded-matrix WMMA ----
    // For a 16-row block rb and 16-col block cb:
    //   D[m][n] = sum_k Band[m][k] * Z[rb+k][cb+n],  Band[m][k]=1 iff m<=k<=m+14
    // computed as 8 chained V_WMMA_F32_16X16X4_F32 (K=0..31; coeffs 0 for k>29).
    // Pad rows/cols 78..79 are zeros, so out-of-window terms are exactly 0.
    for (int t = wave; t < 4 * 5; t += 8) {   // 4 row-blocks x 5 col-blocks
      const int rb = (t / 5) * 16;
      const int cb = (t % 5) * 16;
      const int cc = cb + lm;                 // this lane's column (N = lm)
      v8f acc = {};
#pragma unroll
      for (int k0 = 0; k0 < 32; k0 += 4) {
        const int ka = k0 + 2 * half;         // A: lane holds K = ka, ka+1
        v2f a, bv;
        a.x = (ka     >= lm && ka     <= lm + 14) ? 1.0f : 0.0f;
        a.y = (ka + 1 >= lm && ka + 1 <= lm + 14) ? 1.0f : 0.0f;
        const int rk = rb + ka;               // B: lane holds rows rk, rk+1
        bv.x = s_x[cur][(rk    ) * PITCH + cc] * s_mask[(rk    ) * PITCH + cc];
        bv.y = s_x[cur][(rk + 1) * PITCH + cc] * s_mask[(rk + 1) * PITCH + cc];
        acc = __builtin_amdgcn_wmma_f32_16x16x4_f32(
            false, a, false, bv, (short)0, acc, false, false);
      }
#pragma unroll
      for (int i = 0; i < 8; ++i) {           // D: element i is row m=i+8*half
        s_v[(rb + i + 8 * half) * PITCH + cc] = acc[i];
      }
    }
    __syncthreads();

    // ---- horizontal sliding-window sums + blend -> s_out (256 tasks) ----
    {
      int r  = tid >> 2;
      int cb = (tid & 3) * 16;
      float sum = 0.0f, num = 0.0f;
#pragma unroll
      for (int k = 0; k < WINDOW; ++k) {
        sum += s_v [r * PITCH + cb + k];
        num += s_vn[r * PITCH + cb + k];
      }
#pragma unroll
      for (int j = 0; j < 16; ++j) {
        float mean = (num == 0.0f) ? 0.0f : sum / num;
        float xv = s_x[cur][(r + PADR) * PITCH + cb + j + PADR];
        float mv = s_mask   [(r + PADR) * PITCH + cb + j + PADR];
        s_out[r * OPITCH + cb + j] = xv * mv + mean * (1.0f - mv);
        if (j < 15) {
          sum += s_v [r * PITCH + cb + j + WINDOW] - s_v [r * PITCH + cb + j];
          num += s_vn[r * PITCH + cb + j + WINDOW] - s_vn[r * PITCH + cb + j];
        }
      }
    }
    __syncthreads();

    // ---- coalesced store of the 64x64 tile ----
    float* oc = obase + (size_t)ch * HEIGHT * WIDTH;
    for (int i = tid; i < TILE * TILE; i += NTHREADS) {
      int r = i >> 6, c = i & 63;
      oc[(ty0 + r) * WIDTH + (tx0 + c)] = s_out[r * OPITCH + c];
    }

    // make the prefetched buffer safe to consume next iteration
    wait_async_all();
    __syncthreads();
  }
}

extern "C" void kernel_launch(void* const* d_in, const int* in_sizes, int n_in,
                              void* d_out, int out_size, void* d_ws, size_t ws_size,
                              hipStream_t stream) {
  (void)in_sizes; (void)n_in; (void)out_size; (void)d_ws; (void)ws_size;
  const float* x    = (const float*)d_in[0];
  const float* mask = (const float*)d_in[1];
  // d_in[2] is the all-ones 15x15 kernel; the box window is hardcoded.
  float* out = (float*)d_out;

  dim3 grid(WIDTH / TILE, HEIGHT / TILE, BATCH);
  dim3 block(NTHREADS);
  hipLaunchKernelGGL(inpaint_kernel, grid, block, 0, stream, x, mask, out);
}